// Net_33646773796902
// MI455X (gfx1250) — compile-verified
//
#include <hip/hip_runtime.h>

typedef __attribute__((ext_vector_type(2))) float v2f;
typedef __attribute__((ext_vector_type(8))) float v8f;

#define NBR 5
#define NH  10
#define BN_EPS 1e-5f
#define BLK 256
#define NWAVES 8
#define GRIDX 1024

// ws layout (floats):
//   [0..49]    sum(h1)       [50..99]   sum(h1^2)
//   [100..149] sum(h2)       [150..199] sum(h2^2)
//   [200..249] bn1 scale     [250..299] bn1 shift
//   [300..349] bn2 scale     [350..399] bn2 shift

__device__ __forceinline__ float fast_sigmoid(float x) {
  return __builtin_amdgcn_rcpf(1.0f + __expf(-x));
}
__device__ __forceinline__ float fast_tanh(float x) {
  return 1.0f - 2.0f * __builtin_amdgcn_rcpf(__expf(2.0f * x) + 1.0f);
}
__device__ __forceinline__ v8f wmma_f32_16x16x4(v2f a, v2f b, v8f c) {
  // 8 args: (neg_a, A, neg_b, B, c_mod, C, reuse_a, reuse_b)
  return __builtin_amdgcn_wmma_f32_16x16x4_f32(false, a, false, b, (short)0, c,
                                               false, false);
}

__global__ void zero_stats_kernel(float* ws) {
  int i = threadIdx.x;
  if (i < 200) ws[i] = 0.0f;
}

// Pass 1: h1 = sigmoid(W1*[x,y,t] + b1); accumulate per-feature sum / sumsq.
__global__ __launch_bounds__(BLK) void stats1_kernel(
    const float* __restrict__ x, const float* __restrict__ y,
    const float* __restrict__ t, const float* __restrict__ W1,
    const float* __restrict__ b1, float* __restrict__ ws, int Btot,
    int tilesPerWave) {
  const int k    = blockIdx.y;        // branch
  const int lane = threadIdx.x & 31;
  const int wv   = threadIdx.x >> 5;
  const int hf   = lane >> 4;         // lane half
  const int s    = lane & 15;         // sample column
  const int m    = lane & 15;         // A-matrix row (output feature)

  // A (16x4): V0 = K0 | K2 ; V1 = K1 | K3(pad=0). Rows m>=10 are zero.
  v2f A1 = {0.f, 0.f};
  if (m < NH) {
    const float* w = W1 + (k * NH + m) * 3;
    A1[0] = hf ? w[2] : w[0];
    A1[1] = hf ? 0.0f : w[1];
  }
  v8f C1;
  float accS[8], accQ[8];
#pragma unroll
  for (int r = 0; r < 8; ++r) {
    int mr  = r + 8 * hf;             // C/D row = feature
    C1[r]   = (mr < NH) ? b1[k * NH + mr] : 0.0f;
    accS[r] = 0.f;
    accQ[r] = 0.f;
  }

  const int gw = blockIdx.x * NWAVES + wv;
  for (int i = 0; i < tilesPerWave; ++i) {
    const int base = (gw * tilesPerWave + i) * 16;
    const int idx  = base + s;
    float xv = x[idx], yv = y[idx], tv = t[idx];
    v2f Bm;                           // B (4x16): rows x,y,t,0; cols = samples
    Bm[0] = hf ? tv : xv;
    Bm[1] = hf ? 0.f : yv;
    v8f d = wmma_f32_16x16x4(A1, Bm, C1);
#pragma unroll
    for (int r = 0; r < 8; ++r) {
      float hv = fast_sigmoid(d[r]);
      accS[r] += hv;
      accQ[r] += hv * hv;
    }
  }
  // reduce the 16 sample lanes of each half (masks stay within the half)
#pragma unroll
  for (int r = 0; r < 8; ++r) {
#pragma unroll
    for (int msk = 1; msk < 16; msk <<= 1) {
      accS[r] += __shfl_xor(accS[r], msk, 32);
      accQ[r] += __shfl_xor(accQ[r], msk, 32);
    }
  }
  __shared__ float sS[16], sQ[16];
  if (threadIdx.x < 16) { sS[threadIdx.x] = 0.f; sQ[threadIdx.x] = 0.f; }
  __syncthreads();
  if (s == 0) {                       // lanes 0 and 16 of each wave
#pragma unroll
    for (int r = 0; r < 8; ++r) {
      int mr = r + 8 * hf;
      if (mr < NH) {
        atomicAdd(&sS[mr], accS[r]);
        atomicAdd(&sQ[mr], accQ[r]);
      }
    }
  }
  __syncthreads();
  if (threadIdx.x < NH) {
    atomicAdd(&ws[k * NH + threadIdx.x], sS[threadIdx.x]);
    atomicAdd(&ws[50 + k * NH + threadIdx.x], sQ[threadIdx.x]);
  }
}

__global__ void bn_finalize_kernel(const float* __restrict__ g,
                                   const float* __restrict__ be,
                                   float* __restrict__ ws, int sumOff, int sqOff,
                                   int scOff, int shOff, float invB) {
  int f = threadIdx.x;
  if (f < NBR * NH) {
    float mean = ws[sumOff + f] * invB;
    float var  = ws[sqOff + f] * invB - mean * mean;  // biased, like torch BN
    float sc   = g[f] * rsqrtf(var + BN_EPS);
    ws[scOff + f] = sc;
    ws[shOff + f] = be[f] - mean * sc;
  }
}

// FINAL=false: recompute L1, apply BN1, L2, accumulate h2 stats.
// FINAL=true : full forward, apply BN2, L3 dot + bias (+tanh on branch 3).
template <bool FINAL>
__global__ __launch_bounds__(BLK) void fwd_kernel(
    const float* __restrict__ x, const float* __restrict__ y,
    const float* __restrict__ t, const float* __restrict__ W1,
    const float* __restrict__ b1, const float* __restrict__ W2,
    const float* __restrict__ b2, const float* __restrict__ W3,
    const float* __restrict__ b3, float* __restrict__ ws,
    float* __restrict__ out, int Btot, int tilesPerWave) {
  const int k    = blockIdx.y;
  const int lane = threadIdx.x & 31;
  const int wv   = threadIdx.x >> 5;
  const int hf   = lane >> 4;
  const int s    = lane & 15;
  const int m    = lane & 15;

  v2f A1   = {0.f, 0.f};
  v2f A2g0 = {0.f, 0.f}, A2g1 = {0.f, 0.f}, A2g2 = {0.f, 0.f};
  if (m < NH) {
    const float* w1 = W1 + (k * NH + m) * 3;
    A1[0] = hf ? w1[2] : w1[0];
    A1[1] = hf ? 0.0f : w1[1];
    const float* w2 = W2 + (k * NH + m) * NH;  // row m of W2[k]
    A2g0[0] = hf ? w2[2] : w2[0];   // K = 0..3
    A2g0[1] = hf ? w2[3] : w2[1];
    A2g1[0] = hf ? w2[6] : w2[4];   // K = 4..7
    A2g1[1] = hf ? w2[7] : w2[5];
    A2g2[0] = hf ? 0.0f : w2[8];    // K = 8,9 (cols 10,11 zero-padded)
    A2g2[1] = hf ? 0.0f : w2[9];
  }
  v8f C1, C2;
  float s1v[8], t1v[8], s2v[8], t2v[8], w3v[8];
  float accS[8], accQ[8];
#pragma unroll
  for (int r = 0; r < 8; ++r) {
    int  mr    = r + 8 * hf;
    bool valid = (mr < NH);
    int  f     = k * NH + mr;
    C1[r]  = valid ? b1[f] : 0.0f;
    C2[r]  = valid ? b2[f] : 0.0f;
    s1v[r] = valid ? ws[200 + f] : 0.0f;  // pad features -> 0 (kept finite)
    t1v[r] = valid ? ws[250 + f] : 0.0f;
    if (FINAL) {
      s2v[r] = valid ? ws[300 + f] : 0.0f;
      t2v[r] = valid ? ws[350 + f] : 0.0f;
      w3v[r] = valid ? W3[f] : 0.0f;
    } else {
      s2v[r] = t2v[r] = w3v[r] = 0.0f;
    }
    accS[r] = 0.f;
    accQ[r] = 0.f;
  }
  const float b3k = FINAL ? b3[k] : 0.0f;

  const int gw = blockIdx.x * NWAVES + wv;
  for (int i = 0; i < tilesPerWave; ++i) {
    const int base = (gw * tilesPerWave + i) * 16;
    const int idx  = base + s;
    float xv = x[idx], yv = y[idx], tv = t[idx];
    v2f Bm;
    Bm[0] = hf ? tv : xv;
    Bm[1] = hf ? 0.f : yv;
    v8f d1 = wmma_f32_16x16x4(A1, Bm, C1);
    float H[8];
#pragma unroll
    for (int r = 0; r < 8; ++r)
      H[r] = s1v[r] * fast_sigmoid(d1[r]) + t1v[r];   // BN1 applied; pads -> 0
    // Relayout D(feat x sample) -> B(K x sample) for layer 2, in registers.
    float X0 = __shfl_xor(H[0], 16, 32);
    float X1 = __shfl_xor(H[1], 16, 32);
    float X2 = __shfl_xor(H[2], 16, 32);
    float X3 = __shfl_xor(H[3], 16, 32);
    float X6 = __shfl_xor(H[6], 16, 32);
    float X7 = __shfl_xor(H[7], 16, 32);
    v2f B0, B1v, B2v;
    B0[0]  = hf ? X2 : H[0];  // rows 0 | 2
    B0[1]  = hf ? X3 : H[1];  // rows 1 | 3
    B1v[0] = hf ? X6 : H[4];  // rows 4 | 6
    B1v[1] = hf ? X7 : H[5];  // rows 5 | 7
    B2v[0] = X0;              // row 8 | (row 10 junk * zero A col)
    B2v[1] = X1;              // row 9 | (row 11 junk * zero A col)
    v8f d2 = wmma_f32_16x16x4(A2g0, B0, C2);
    d2     = wmma_f32_16x16x4(A2g1, B1v, d2);
    d2     = wmma_f32_16x16x4(A2g2, B2v, d2);
    if (!FINAL) {
#pragma unroll
      for (int r = 0; r < 8; ++r) {
        float hv = fast_sigmoid(d2[r]);
        accS[r] += hv;
        accQ[r] += hv * hv;
      }
    } else {
      float p = 0.f;
#pragma unroll
      for (int r = 0; r < 8; ++r)
        p = fmaf(w3v[r], s2v[r] * fast_sigmoid(d2[r]) + t2v[r], p);
      p += __shfl_xor(p, 16, 32);  // combine feature halves per sample
      if (hf == 0) {
        float o = p + b3k;
        if (k == 3) o = fast_tanh(o);   // phi branch
        out[k * Btot + idx] = o;
      }
    }
  }

  if (!FINAL) {
#pragma unroll
    for (int r = 0; r < 8; ++r) {
#pragma unroll
      for (int msk = 1; msk < 16; msk <<= 1) {
        accS[r] += __shfl_xor(accS[r], msk, 32);
        accQ[r] += __shfl_xor(accQ[r], msk, 32);
      }
    }
    __shared__ float sS[16], sQ[16];
    if (threadIdx.x < 16) { sS[threadIdx.x] = 0.f; sQ[threadIdx.x] = 0.f; }
    __syncthreads();
    if (s == 0) {
#pragma unroll
      for (int r = 0; r < 8; ++r) {
        int mr = r + 8 * hf;
        if (mr < NH) {
          atomicAdd(&sS[mr], accS[r]);
          atomicAdd(&sQ[mr], accQ[r]);
        }
      }
    }
    __syncthreads();
    if (threadIdx.x < NH) {
      atomicAdd(&ws[100 + k * NH + threadIdx.x], sS[threadIdx.x]);
      atomicAdd(&ws[150 + k * NH + threadIdx.x], sQ[threadIdx.x]);
    }
  }
}

extern "C" void kernel_launch(void* const* d_in, const int* in_sizes, int n_in,
                              void* d_out, int out_size, void* d_ws,
                              size_t ws_size, hipStream_t stream) {
  const float* x   = (const float*)d_in[0];
  const float* y   = (const float*)d_in[1];
  const float* t   = (const float*)d_in[2];
  const float* W1  = (const float*)d_in[3];
  const float* b1  = (const float*)d_in[4];
  const float* g1  = (const float*)d_in[5];
  const float* be1 = (const float*)d_in[6];
  const float* W2  = (const float*)d_in[7];
  const float* b2  = (const float*)d_in[8];
  const float* g2  = (const float*)d_in[9];
  const float* be2 = (const float*)d_in[10];
  const float* W3  = (const float*)d_in[11];
  const float* b3  = (const float*)d_in[12];
  float* ws  = (float*)d_ws;
  float* out = (float*)d_out;
  const int   Btot = in_sizes[0];                    // 2097152
  const int   tilesPerWave = Btot / (GRIDX * NWAVES * 16);  // 16
  const float invB = 1.0f / (float)Btot;
  dim3 grid(GRIDX, NBR);

  zero_stats_kernel<<<1, 256, 0, stream>>>(ws);
  stats1_kernel<<<grid, BLK, 0, stream>>>(x, y, t, W1, b1, ws, Btot,
                                          tilesPerWave);
  bn_finalize_kernel<<<1, 64, 0, stream>>>(g1, be1, ws, 0, 50, 200, 250, invB);
  fwd_kernel<false><<<grid, BLK, 0, stream>>>(x, y, t, W1, b1, W2, b2, W3, b3,
                                              ws, out, Btot, tilesPerWave);
  bn_finalize_kernel<<<1, 64, 0, stream>>>(g2, be2, ws, 100, 150, 300, 350,
                                           invB);
  fwd_kernel<true><<<grid, BLK, 0, stream>>>(x, y, t, W1, b1, W2, b2, W3, b3,
                                             ws, out, Btot, tilesPerWave);
}